// TensorAggregateLayer_3968549782327
// MI455X (gfx1250) — compile-verified
//
#include <hip/hip_runtime.h>
#include <hip/hip_bf16.h>
#include <math.h>

// Problem constants from the reference
#define B_   8
#define N_   4096
#define M_   64
#define C_   128
#define K_   8
#define PI_  3.14159265358979323846f
#define RCUT 5.0f

typedef __attribute__((ext_vector_type(2))) float v2f;
typedef __attribute__((ext_vector_type(8))) float v8f;

// One workgroup per (b,n). 256 threads = 8 wave32s.
//  Phase A: 64 threads (2 waves) compute neighbor moments s, v[3], S[3][3].
//  Phase B: build 16x16 padded map matrix M, stage 128x13 feature tile U in LDS,
//           each wave computes a 16-channel tile via 4x V_WMMA_F32_16X16X4_F32:
//           Y(16x16) = U(16x16[13 used]) x B(16x16), B[p][q] = M[q][p].
__global__ __launch_bounds__(256) void tensor_agg_kernel(
    const float* __restrict__ t0,      // (B,N,C)
    const float* __restrict__ t1,      // (B,N,C,3)
    const float* __restrict__ t2,      // (B,N,C,3,3)
    const float* __restrict__ coord,   // (B,N,3)
    const float* __restrict__ rw,      // (K,)
    const int*   __restrict__ nbr,     // (B,N,M)
    const unsigned char* __restrict__ msk, // (B,N,M) bool
    float* __restrict__ out)           // out0 | out1 | out2 concatenated
{
  __shared__ float Ulds[C_][17];   // 17-float pitch: bank-conflict padding
  __shared__ float Mlds[16][17];
  __shared__ float redS[13];       // s, v[3], S[3][3] (row-major)

  const int tid = threadIdx.x;
  const int bn  = blockIdx.x;              // b*N + n
  const int b   = bn >> 12;                // N_ == 4096
  const int n   = bn & (N_ - 1);

  if (tid < 13) redS[tid] = 0.0f;
  __syncthreads();

  // ---------------- Phase A: neighbor moments ----------------
  if (tid < M_) {
    const size_t cb = (size_t)b * N_ * 3;
    const float cx = coord[cb + (size_t)n * 3 + 0];
    const float cy = coord[cb + (size_t)n * 3 + 1];
    const float cz = coord[cb + (size_t)n * 3 + 2];
    const int j = nbr[(size_t)bn * M_ + tid];
    const float rx = coord[cb + (size_t)j * 3 + 0] - cx;
    const float ry = coord[cb + (size_t)j * 3 + 1] - cy;
    const float rz = coord[cb + (size_t)j * 3 + 2] - cz;
    const float d  = sqrtf(rx * rx + ry * ry + rz * rz + 1e-12f);

    const float arg = d * (PI_ / RCUT);
    float basis = 0.0f;
#pragma unroll
    for (int k = 0; k < K_; ++k)
      basis += __sinf((float)(k + 1) * arg) * rw[k];
    const float xr  = fminf(d * (1.0f / RCUT), 1.0f);
    const float cut = 0.5f * (__cosf(PI_ * xr) + 1.0f);
    float fn = basis * cut;
    if (!msk[(size_t)bn * M_ + tid]) fn = 0.0f;

    const float r3[3] = {rx, ry, rz};
    float vals[13];
    vals[0] = fn;
    vals[1] = fn * rx; vals[2] = fn * ry; vals[3] = fn * rz;
#pragma unroll
    for (int x = 0; x < 3; ++x)
#pragma unroll
      for (int y = 0; y < 3; ++y)
        vals[4 + 3 * x + y] = fn * r3[x] * r3[y];

    // wave32 butterfly reduce, then one atomic set per wave (2 waves active)
#pragma unroll
    for (int off = 16; off > 0; off >>= 1)
#pragma unroll
      for (int q = 0; q < 13; ++q)
        vals[q] += __shfl_xor(vals[q], off, 32);
    if ((tid & 31) == 0)
#pragma unroll
      for (int q = 0; q < 13; ++q)
        atomicAdd(&redS[q], vals[q]);
  }
  __syncthreads();

  // ---------------- Build padded 16x16 map matrix M[q][p] ----------------
  {
    const int q = tid >> 4;          // output component 0..15
    const int p = tid & 15;          // input component 0..15
    float val = 0.0f;
    if (p < 13 && q < 13) {
      const float s = redS[0];
      if (q == 0) {
        val = redS[p];                               // s | v | S (flat)
      } else if (q < 4) {
        const int x = q - 1;
        if (p == 0)       val = redS[1 + x];                                  // v[x]
        else if (p < 4) { const int i = p - 1;
                          val = redS[4 + 3 * i + x] + (i == x ? s : 0.0f); }  // S[i,x] + s*delta
        else            { const int pp = p - 4, xp = pp / 3, jj = pp - 3 * xp;
                          val = (xp == x) ? redS[1 + jj] : 0.0f; }            // v[j]
      } else {
        const int qq = q - 4, x = qq / 3, yy = qq - 3 * x;
        if (p == 0)       val = redS[q];                                      // S[x,yy]
        else if (p < 4) { const int i = p - 1;
                          val = (i == x) ? redS[1 + yy] : 0.0f; }             // v[yy]
        else            { const int pp = p - 4, xp = pp / 3, jj = pp - 3 * xp;
                          val = (xp == x)
                              ? (redS[4 + 3 * jj + yy] + (jj == yy ? s : 0.0f))
                              : 0.0f; }                                       // S[j,yy] + s*delta
      }
    }
    Mlds[q][p] = val;
  }

  // ---------------- Stage feature tile U (coalesced) ----------------
  const size_t baseC = (size_t)bn * C_;
  for (int idx = tid; idx < C_;     idx += 256) Ulds[idx][0]               = t0[baseC + idx];
  for (int idx = tid; idx < C_ * 3; idx += 256) Ulds[idx / 3][1 + idx % 3] = t1[baseC * 3 + idx];
  for (int idx = tid; idx < C_ * 9; idx += 256) Ulds[idx / 9][4 + idx % 9] = t2[baseC * 9 + idx];
  for (int idx = tid; idx < C_ * 4; idx += 256) Ulds[idx >> 2][13 + (idx & 3)] = 0.0f; // K pad
  __syncthreads();

  // ---------------- Phase B: WMMA f32 16x16x4, K chained 0..15 ----------------
  const int wave = tid >> 5;
  const int lane = tid & 31;
  const int row  = (wave << 4) + (lane & 15);   // channel for A fragment
  const int kh   = (lane >> 4) << 1;            // K sub-pair: lanes 16-31 hold K+2,K+3
  const int q    = lane & 15;                   // output column for B fragment

  v8f acc = {0.f, 0.f, 0.f, 0.f, 0.f, 0.f, 0.f, 0.f};
#pragma unroll
  for (int k0 = 0; k0 < 16; k0 += 4) {
    v2f a, bb;
    a.x  = Ulds[row][k0 + kh];       // A[m=row, k]
    a.y  = Ulds[row][k0 + kh + 1];
    bb.x = Mlds[q][k0 + kh];         // B[k, q] = M[q][k]
    bb.y = Mlds[q][k0 + kh + 1];
    acc = __builtin_amdgcn_wmma_f32_16x16x4_f32(
        /*neg_a=*/false, a, /*neg_b=*/false, bb,
        /*c_mod=*/(short)0, acc, /*reuse_a=*/false, /*reuse_b=*/false);
  }

  // ---------------- Branchless scatter of the 13 valid output columns ----
  // q==0 -> out0, stride 1, sub 0
  // q<4  -> out1, stride 3, sub q-1
  // else -> out2, stride 9, sub q-4
  const size_t BNC = (size_t)B_ * N_ * C_;
  const int    mult  = (q == 0) ? 1 : ((q < 4) ? 3 : 9);
  const size_t obase = (q == 0) ? (size_t)0 : ((q < 4) ? BNC : BNC * 4);
  const int    sub   = (q == 0) ? 0 : ((q < 4) ? (q - 1) : (q - 4));
  float* dst = out + obase + baseC * (size_t)mult + (size_t)sub;

  const int rbase = (lane >> 4) << 3;             // lanes 16-31 hold rows m+8
  const int cbase = (wave << 4) + rbase;
  if (q < 13) {
#pragma unroll
    for (int r = 0; r < 8; ++r)
      dst[(size_t)(cbase + r) * (size_t)mult] = acc[r];
  }
}

extern "C" void kernel_launch(void* const* d_in, const int* in_sizes, int n_in,
                              void* d_out, int out_size, void* d_ws, size_t ws_size,
                              hipStream_t stream) {
  const float* t0    = (const float*)d_in[0];
  const float* t1    = (const float*)d_in[1];
  const float* t2    = (const float*)d_in[2];
  const float* coord = (const float*)d_in[3];
  const float* rw    = (const float*)d_in[4];
  const int*   nbr   = (const int*)d_in[5];
  const unsigned char* msk = (const unsigned char*)d_in[6];
  float* out = (float*)d_out;

  dim3 grid(B_ * N_);   // 32768 workgroups, one per (b,n)
  dim3 block(256);      // 8 wave32s
  tensor_agg_kernel<<<grid, block, 0, stream>>>(t0, t1, t2, coord, rw, nbr, msk, out);

  (void)in_sizes; (void)n_in; (void)out_size; (void)d_ws; (void)ws_size;
}